// GCNDecoder_66125316489524
// MI455X (gfx1250) — compile-verified
//
#include <hip/hip_runtime.h>
#include <math.h>

typedef float v2f __attribute__((ext_vector_type(2)));
typedef float v8f __attribute__((ext_vector_type(8)));

#define IN_CH 128
#define HID   128
#define OUT_CH 64

static inline int div_up(long long a, long long b) { return (int)((a + b - 1) / b); }

// ---------------------------------------------------------------------------
// Degree / normalization
// ---------------------------------------------------------------------------
__global__ void k_deg_init(float* __restrict__ deg, int n) {
    int i = blockIdx.x * blockDim.x + threadIdx.x;
    if (i < n) deg[i] = 1.0f;  // self-loop contributes 1
}

__global__ void k_deg_scatter(const int* __restrict__ dst, float* __restrict__ deg, int nE) {
    int e = blockIdx.x * blockDim.x + threadIdx.x;
    if (e < nE) atomicAdd(&deg[dst[e]], 1.0f);
}

__global__ void k_dinv(float* __restrict__ deg, int n) {
    int i = blockIdx.x * blockDim.x + threadIdx.x;
    if (i < n) {
        float d = deg[i];
        deg[i] = (d > 0.0f) ? rsqrtf(d) : 0.0f;  // d >= 1 always (self-loop)
    }
}

// ---------------------------------------------------------------------------
// f32 WMMA GEMM: out[nrows, FOUT] = A[nrows, K] @ W[K, FOUT]
// One wave per 16x16 output tile (col tile = wave id). W staged in LDS.
// A-frag layout (16x4 f32, ISA 7.12.2): lanes 0-15 = rows M=0..15 holding
// K = {k0, k0+1}; lanes 16-31 same rows holding K = {k0+2, k0+3}.
// B-frag: lane halves mirror the K split, N = lane&15 within each half.
// C/D: VGPR r -> row (half*8 + r), col = lane&15.
// ---------------------------------------------------------------------------
template <int K, int FOUT>
__global__ void k_gemm_wmma_f32(const float* __restrict__ A,
                                const float* __restrict__ W,
                                float* __restrict__ out, int nrows) {
    __shared__ float sW[K * FOUT];
    const int tid = threadIdx.x;
    for (int i = tid; i < K * FOUT; i += blockDim.x) sW[i] = W[i];
    __syncthreads();

    const int wave = tid >> 5;
    const int lane = tid & 31;
    const int half = lane >> 4;      // 0: K pair {0,1}; 1: K pair {2,3}
    const int l16  = lane & 15;
    const int rowBase = blockIdx.x * 16;
    const int colBase = wave * 16;
    if (rowBase >= nrows) return;

    const float* __restrict__ arow = A + (size_t)(rowBase + l16) * K;

    v8f c = {0.f, 0.f, 0.f, 0.f, 0.f, 0.f, 0.f, 0.f};
    for (int k0 = 0; k0 < K; k0 += 4) {
        // contiguous K pair per lane -> single global_load_b64
        v2f a = *(const v2f*)(arow + k0 + half * 2);
        v2f b;
        b.x = sW[(k0 + half * 2 + 0) * FOUT + colBase + l16];
        b.y = sW[(k0 + half * 2 + 1) * FOUT + colBase + l16];
        c = __builtin_amdgcn_wmma_f32_16x16x4_f32(
                /*neg_a=*/false, a, /*neg_b=*/false, b,
                /*c_mod=*/(short)0, c, /*reuse_a=*/false, /*reuse_b=*/false);
    }

#pragma unroll
    for (int r = 0; r < 8; ++r) {
        int row = rowBase + half * 8 + r;
        out[(size_t)row * FOUT + colBase + l16] = c[r];
    }
}

// ---------------------------------------------------------------------------
// agg[i,f] = h[i,f] * dinv[i]^2 + bias[f]   (self-loop term + bias init)
// ---------------------------------------------------------------------------
template <int F>
__global__ void k_selfloop_bias(const float* __restrict__ h,
                                const float* __restrict__ dinv,
                                const float* __restrict__ bias,
                                float* __restrict__ agg, int n) {
    long long i = (long long)blockIdx.x * blockDim.x + threadIdx.x;
    if (i >= (long long)n * F) return;
    int node = (int)(i / F);
    int f    = (int)(i % F);
    float d = dinv[node];
    agg[i] = h[i] * d * d + bias[f];
}

// ---------------------------------------------------------------------------
// Edge scatter: one wave per edge; lanes sweep F features in 32-wide
// coalesced bursts of global_atomic_add_f32.
// ---------------------------------------------------------------------------
template <int F>
__global__ void k_edge_scatter(const int* __restrict__ src,
                               const int* __restrict__ dst,
                               const float* __restrict__ h,
                               const float* __restrict__ dinv,
                               float* __restrict__ agg, int nE) {
    int gw   = (blockIdx.x * blockDim.x + threadIdx.x) >> 5;
    int lane = threadIdx.x & 31;
    if (gw >= nE) return;
    int s = src[gw];
    int d = dst[gw];
    float norm = dinv[s] * dinv[d];
    const float* __restrict__ hs = h + (size_t)s * F;
    float* __restrict__ ad = agg + (size_t)d * F;
#pragma unroll
    for (int c = 0; c < F / 32; ++c) {
        int f = lane + 32 * c;
        atomicAdd(&ad[f], hs[f] * norm);
    }
}

__global__ void k_tanh(float* __restrict__ a, long long n) {
    long long i = (long long)blockIdx.x * blockDim.x + threadIdx.x;
    if (i < n) a[i] = tanhf(a[i]);
}

// ---------------------------------------------------------------------------
// Launch
// ---------------------------------------------------------------------------
extern "C" void kernel_launch(void* const* d_in, const int* in_sizes, int n_in,
                              void* d_out, int out_size, void* d_ws, size_t ws_size,
                              hipStream_t stream) {
    const float* x  = (const float*)d_in[0];
    const int*   ei = (const int*)d_in[1];   // [2, E]: row 0 = src, row 1 = dst
    const float* W1 = (const float*)d_in[2];
    const float* b1 = (const float*)d_in[3];
    const float* W2 = (const float*)d_in[4];
    const float* b2 = (const float*)d_in[5];

    const int nN = in_sizes[0] / IN_CH;      // 50000
    const int nE = in_sizes[1] / 2;          // 800000
    const int* src = ei;
    const int* dst = ei + nE;

    float* ws   = (float*)d_ws;
    float* dinv = ws;                                     // nN floats
    float* hbuf = ws + (((size_t)nN + 63) & ~(size_t)63); // nN*HID (reused as h2: nN*OUT_CH)
    float* abuf = hbuf + (size_t)nN * HID;                // nN*HID
    float* out  = (float*)d_out;                          // nN*OUT_CH

    // 1. symmetric normalization coefficients
    k_deg_init<<<div_up(nN, 256), 256, 0, stream>>>(dinv, nN);
    k_deg_scatter<<<div_up(nE, 256), 256, 0, stream>>>(dst, dinv, nE);
    k_dinv<<<div_up(nN, 256), 256, 0, stream>>>(dinv, nN);

    // 2. layer 1: h1 = x @ W1  (WMMA f32), then aggregate, then tanh
    k_gemm_wmma_f32<IN_CH, HID>
        <<<div_up(nN, 16), (HID / 16) * 32, 0, stream>>>(x, W1, hbuf, nN);
    k_selfloop_bias<HID>
        <<<div_up((long long)nN * HID, 256), 256, 0, stream>>>(hbuf, dinv, b1, abuf, nN);
    k_edge_scatter<HID>
        <<<div_up((long long)nE * 32, 256), 256, 0, stream>>>(src, dst, hbuf, dinv, abuf, nE);
    k_tanh<<<div_up((long long)nN * HID, 256), 256, 0, stream>>>(abuf, (long long)nN * HID);

    // 3. layer 2: h2 = a1 @ W2 (WMMA f32), aggregate directly into d_out
    k_gemm_wmma_f32<HID, OUT_CH>
        <<<div_up(nN, 16), (OUT_CH / 16) * 32, 0, stream>>>(abuf, W2, hbuf, nN);
    k_selfloop_bias<OUT_CH>
        <<<div_up((long long)nN * OUT_CH, 256), 256, 0, stream>>>(hbuf, dinv, b2, out, nN);
    k_edge_scatter<OUT_CH>
        <<<div_up((long long)nE * 32, 256), 256, 0, stream>>>(src, dst, hbuf, dinv, out, nE);
}